// DCROutputs_69767448756596
// MI455X (gfx1250) — compile-verified
//
#include <hip/hip_runtime.h>
#include <hip/hip_bf16.h>

// Problem constants (fixed by the reference: pred_disp [16, 2, 768, 768] f32, NUM_IT=4)
#define N_ 16
#define H_ 768
#define W_ 768
constexpr int HW  = H_ * W_;        // 589,824
constexpr int NHW = N_ * HW;        // 9,437,184 pixels
constexpr int THREADS = 256;        // 8 wave32s per block; W = 3 * 256 exactly

// Grid for all iteration kernels: (W/256, H, N). n and y are scalar (SGPR),
// so plane/row bases are scalar and coalesced loads use the saddr form; no
// integer division, no tail predication.

// ---------------------------------------------------------------------------
// Iteration 0: planar input [N,2,H,W] -> interleaved float2 buffer [N,H,W]
// ---------------------------------------------------------------------------
__global__ __launch_bounds__(THREADS) void dcr_iter_first(
    const float* __restrict__ in,      // [N,2,H,W] planar
    float2* __restrict__ dst) {        // [N,H,W] interleaved
  const int x = blockIdx.x * THREADS + threadIdx.x;   // 0..767, exact
  const int y = blockIdx.y;                           // scalar
  const int n = blockIdx.z;                           // scalar
  const int p = y * W_ + x;

  const float* __restrict__ plane = in + (size_t)n * (2 * HW);
  const float dx = plane[p];          // coalesced, scalar base
  const float dy = plane[HW + p];

  int cx = (int)((float)x + dx);      // trunc toward zero == .astype(int32)
  int cy = (int)((float)y + dy);
  cx = min(max(cx, 0), W_ - 1);
  cy = min(max(cy, 0), H_ - 1);

  const int gp = cy * W_ + cx;
  const float gx = plane[gp];         // random gather; L2-resident source
  const float gy = plane[HW + gp];

  dst[(size_t)n * HW + p] = make_float2(gx + dx, gy + dy);
}

// ---------------------------------------------------------------------------
// Iterations 1..2: interleaved -> interleaved. Gather is one b64 load.
// ---------------------------------------------------------------------------
__global__ __launch_bounds__(THREADS) void dcr_iter_mid(
    const float2* __restrict__ src,    // [N,H,W] interleaved
    float2* __restrict__ dst) {        // [N,H,W] interleaved
  const int x = blockIdx.x * THREADS + threadIdx.x;
  const int y = blockIdx.y;
  const int n = blockIdx.z;
  const int p = y * W_ + x;

  const float2* __restrict__ plane = src + (size_t)n * HW;
  const float2 d = plane[p];          // coalesced global_load_b64, scalar base

  int cx = (int)((float)x + d.x);
  int cy = (int)((float)y + d.y);
  cx = min(max(cx, 0), W_ - 1);
  cy = min(max(cy, 0), H_ - 1);

  const float2 g = plane[cy * W_ + cx];   // single random b64 gather (L2 hit)

  dst[(size_t)n * HW + p] = make_float2(g.x + d.x, g.y + d.y);
}

// ---------------------------------------------------------------------------
// Iteration 3 (last): interleaved -> planar disp output, plus num_touch
// scatter-add and pred_cent (b, cx, cy), all from the same last-iter cent.
// ---------------------------------------------------------------------------
__global__ __launch_bounds__(THREADS) void dcr_iter_last(
    const float2* __restrict__ src,    // [N,H,W] interleaved
    float* __restrict__ out_disp,      // [N,2,H,W] planar
    int* __restrict__ num_touch,       // [N,H,W], pre-zeroed
    int* __restrict__ pred_cent) {     // [N,3,H,W]
  const int x = blockIdx.x * THREADS + threadIdx.x;
  const int y = blockIdx.y;
  const int n = blockIdx.z;
  const int p = y * W_ + x;

  const float2* __restrict__ plane = src + (size_t)n * HW;
  const float2 d = plane[p];

  int cx = (int)((float)x + d.x);
  int cy = (int)((float)y + d.y);
  cx = min(max(cx, 0), W_ - 1);
  cy = min(max(cy, 0), H_ - 1);

  const float2 g = plane[cy * W_ + cx];

  float* __restrict__ dplane = out_disp + (size_t)n * (2 * HW);
  dplane[p]      = g.x + d.x;
  dplane[HW + p] = g.y + d.y;

  // scatter-add touch count at (n, cy, cx)
  atomicAdd(num_touch + (size_t)n * HW + cy * W_ + cx, 1);

  // pred_cent: ch0 = batch idx, ch1 = cx, ch2 = cy. Write-once 113MB stream
  // -> non-temporal stores so it doesn't evict the L2-resident disp buffer.
  int* __restrict__ cplane = pred_cent + (size_t)n * (3 * HW);
  __builtin_nontemporal_store(n,  cplane + p);
  __builtin_nontemporal_store(cx, cplane + HW + p);
  __builtin_nontemporal_store(cy, cplane + 2 * HW + p);
}

// ---------------------------------------------------------------------------
// num_touch zero-init (deterministic per call; graph-capture safe)
// ---------------------------------------------------------------------------
__global__ __launch_bounds__(THREADS) void dcr_zero_i32(int* __restrict__ q, int n) {
  int i = blockIdx.x * blockDim.x + threadIdx.x;
  if (i < n) q[i] = 0;
}

extern "C" void kernel_launch(void* const* d_in, const int* in_sizes, int n_in,
                              void* d_out, int out_size, void* d_ws, size_t ws_size,
                              hipStream_t stream) {
  (void)in_sizes; (void)n_in; (void)out_size; (void)ws_size;

  const float* in = (const float*)d_in[0];          // [16,2,768,768] f32

  // d_out layout (concatenated flat, return order):
  //   disp      [N,2,H,W] f32   -> 2*NHW floats
  //   num_touch [N,H,W]   i32   -> NHW ints
  //   pred_cent [N,3,H,W] i32   -> 3*NHW ints
  float* out_disp  = (float*)d_out;
  int*   num_touch = (int*)((float*)d_out + (size_t)2 * NHW);
  int*   pred_cent = num_touch + (size_t)NHW;

  // Double-buffer: ws (interleaved float2) <-> d_out's disp region (same size:
  // N*2*H*W floats == N*H*W float2). Final kernel overwrites d_out planar.
  float2* bufA = (float2*)d_ws;                     // needs NHW*8 = 75.5 MB
  float2* bufB = (float2*)d_out;                    // reused as interleaved temp

  const dim3 grid(W_ / THREADS, H_, N_);            // (3, 768, 16)
  const dim3 block(THREADS);

  dcr_zero_i32<<<(NHW + THREADS - 1) / THREADS, THREADS, 0, stream>>>(num_touch, NHW);

  // NUM_IT = 4 iterations with a grid-wide dependency between each.
  dcr_iter_first<<<grid, block, 0, stream>>>(in, bufA);       // it 0
  dcr_iter_mid  <<<grid, block, 0, stream>>>(bufA, bufB);     // it 1
  dcr_iter_mid  <<<grid, block, 0, stream>>>(bufB, bufA);     // it 2
  dcr_iter_last <<<grid, block, 0, stream>>>(bufA, out_disp,  // it 3
                                             num_touch, pred_cent);
}